// OrthogonalLayer1D_44736379355397
// MI455X (gfx1250) — compile-verified
//
#include <hip/hip_runtime.h>

typedef __attribute__((ext_vector_type(2))) float v2f;
typedef __attribute__((ext_vector_type(4))) float v4f;
typedef __attribute__((ext_vector_type(8))) float v8f;

#define NMODELS 8
#define NSAMP   65536
#define DIM     256
#define XPITCH  260          // DIM + 4 pad floats: LDS row stride (bank-conflict-free strided reads)
#define WPITCH  18           // 16x16 tile pitch, even -> 8B-aligned v2f reads
#define WAVE_LDS (16 * XPITCH + 16 * WPITCH + 16 * WPITCH)

// One wave32 = two samples. rows 0-7 of the 16x256 tile = models of s0, rows 8-15 = s1.
__global__ __launch_bounds__(64)
void cholqr_gs_kernel(const float* __restrict__ x, float* __restrict__ out)
{
    __shared__ float lds[2 * WAVE_LDS];
    const int lane = threadIdx.x & 31;
    const int wib  = threadIdx.x >> 5;
    float* xs = lds + wib * WAVE_LDS;      // 16 x XPITCH staged X tile
    float* gs = xs + 16 * XPITCH;          // 16 x WPITCH Gram matrix
    float* ws = gs + 16 * WPITCH;          // 16 x WPITCH block-diag W = L^-1

    const int  pair = blockIdx.x * 2 + wib;
    const long s0   = (long)pair * 2;
    const long s1   = s0 + 1;

    // ---- stage X into LDS: 16 rows x 64 float4 = 1024 vec4 loads, 32 lanes
    #pragma unroll 4
    for (int i = 0; i < 32; ++i) {
        int  f  = i * 32 + lane;           // float4 id 0..1023
        int  r  = f >> 6;                  // tile row 0..15
        int  c4 = f & 63;                  // float4 within row
        int  m  = r & 7;                   // model index
        long s  = (r < 8) ? s0 : s1;
        const v4f* src = (const v4f*)(x + ((long)m * NSAMP + s) * DIM) + c4;
        v4f v = __builtin_nontemporal_load(src);
        *(v4f*)(xs + r * XPITCH + c4 * 4) = v;
    }
    __syncthreads();

    const int hi  = lane >> 4;             // half-wave select
    const int l15 = lane & 15;

    // ---- phase 1: G16 = Xt * Xt^T via 64 accumulating f32 WMMAs.
    // A layout (16x4 f32): lane<16 holds (M=lane, K=4c+0..1), lane>=16 (M=lane-16, K=4c+2..3).
    // The mirrored B layout means the same registers serve as B = A^T.
    v8f g = {};
    #pragma unroll 8
    for (int c = 0; c < 64; ++c) {
        v2f a = *(const v2f*)(xs + l15 * XPITCH + 4 * c + 2 * hi);
        g = __builtin_amdgcn_wmma_f32_16x16x4_f32(false, a, false, a, (short)0, g,
                                                  false, false);
    }
    // C/D layout: vgpr r -> (M = r + 8*hi, N = l15)
    #pragma unroll
    for (int r = 0; r < 8; ++r)
        gs[(hi * 8 + r) * WPITCH + l15] = g[r];
    __syncthreads();

    // ---- phase 2: per-sample 8x8 Cholesky (G = L L^T) and W = L^-1.
    // Lane 0 handles s0, lane 16 handles s1; fully unrolled -> stays in VGPRs.
    if (l15 == 0) {
        const float* G = gs + hi * (8 * WPITCH + 8);   // diagonal 8x8 block
        float Lm[8][8], W[8][8], invd[8];
        #pragma unroll
        for (int j = 0; j < 8; ++j) {
            float d = G[j * WPITCH + j];
            #pragma unroll
            for (int k = 0; k < j; ++k) d -= Lm[j][k] * Lm[j][k];
            float dj = sqrtf(fmaxf(d, 1e-30f));
            Lm[j][j] = dj;
            invd[j]  = 1.0f / dj;
            #pragma unroll
            for (int i2 = j + 1; i2 < 8; ++i2) {
                float acc = G[i2 * WPITCH + j];
                #pragma unroll
                for (int k = 0; k < j; ++k) acc -= Lm[i2][k] * Lm[j][k];
                Lm[i2][j] = acc * invd[j];
            }
        }
        #pragma unroll
        for (int j = 0; j < 8; ++j) {
            W[j][j] = invd[j];
            #pragma unroll
            for (int i2 = j + 1; i2 < 8; ++i2) {
                float acc = 0.0f;
                #pragma unroll
                for (int k = j; k < i2; ++k) acc += Lm[i2][k] * W[k][j];
                W[i2][j] = -acc * invd[i2];
            }
        }
        // write block-diagonal W16 (constant-index W accesses; hi folded into cndmask)
        #pragma unroll
        for (int r = 0; r < 8; ++r) {
            #pragma unroll
            for (int cc = 0; cc < 16; ++cc) {
                float v = (((cc >> 3) == hi) && ((cc & 7) <= r)) ? W[r][cc & 7] : 0.0f;
                ws[(hi * 8 + r) * WPITCH + cc] = v;
            }
        }
    }
    __syncthreads();

    // ---- phase 3: Q = W16 * Xt (16x16 @ 16x256): 16 col-tiles x 4 K-chunks.
    v2f aw[4];
    #pragma unroll
    for (int c = 0; c < 4; ++c)
        aw[c] = *(const v2f*)(ws + l15 * WPITCH + 4 * c + 2 * hi);

    const long sBase = (hi ? s1 : s0) * DIM;   // lanes<16 -> s0, lanes>=16 -> s1

    #pragma unroll 2
    for (int t = 0; t < 16; ++t) {
        const int n0 = t * 16;
        v8f q = {};
        #pragma unroll
        for (int c = 0; c < 4; ++c) {
            int rb = 4 * c + 2 * hi;           // B rows: K = rb, rb+1
            v2f b;
            b.x = xs[(rb + 0) * XPITCH + n0 + l15];
            b.y = xs[(rb + 1) * XPITCH + n0 + l15];
            q = __builtin_amdgcn_wmma_f32_16x16x4_f32(false, aw[c], false, b, (short)0,
                                                      q, false, false);
        }
        // D layout: vgpr r -> model r, sample (hi ? s1 : s0), col n0+l15
        #pragma unroll
        for (int r = 0; r < 8; ++r) {
            float* dst = out + (long)r * NSAMP * DIM + sBase + n0 + l15;
            __builtin_nontemporal_store(q[r], dst);
        }
    }
}

extern "C" void kernel_launch(void* const* d_in, const int* in_sizes, int n_in,
                              void* d_out, int out_size, void* d_ws, size_t ws_size,
                              hipStream_t stream) {
    (void)in_sizes; (void)n_in; (void)out_size; (void)d_ws; (void)ws_size;
    const float* x   = (const float*)d_in[0];
    float*       out = (float*)d_out;
    // 65536 samples / (2 samples per wave * 2 waves per block) = 16384 blocks
    dim3 grid(NSAMP / 4), block(64);
    cholqr_gs_kernel<<<grid, block, 0, stream>>>(x, out);
}